// Interaction_8297876816075
// MI455X (gfx1250) — compile-verified
//
#include <hip/hip_runtime.h>

// MI455X / gfx1250, wave32. FP32 WMMA path (V_WMMA_F32_16X16X4_F32).
typedef float v2f __attribute__((ext_vector_type(2)));
typedef float v4f __attribute__((ext_vector_type(4)));
typedef float v8f __attribute__((ext_vector_type(8)));

#define NB 16
#define L  256
#define DK 64

// ---------------------------------------------------------------------------
// Kernel A: fold the conv weights.
//   S_up[di,dj,ci]        = sum_co W_up[di,dj,ci,co]
//   W2[rq,rk,di,dj,c]     = (1/16) * sum_ci W_down[di,dj,c,ci] * S_up[3-rq,3-rk,ci]
//   cbias[rq,rk]          = (1/16) * ( b_down . S_up[3-rq,3-rk,:] + sum(b_up) )
// (1/16 = 0.5 mix weight / TEMPERATURE)
// ---------------------------------------------------------------------------
__global__ __launch_bounds__(256) void prep_weights(
    const float* __restrict__ Wd, const float* __restrict__ bd,
    const float* __restrict__ Wu, const float* __restrict__ bu,
    float* __restrict__ W2, float* __restrict__ cbias)
{
  __shared__ float sup[1024];                 // [di*4+dj][ci]
  const int t = threadIdx.x;
  #pragma unroll
  for (int it = 0; it < 4; ++it) {
    const int e = t + it * 256;               // e = (di*4+dj)*64 + ci
    const float* src = Wu + e * 64;
    float s = 0.f;
    for (int co = 0; co < 64; ++co) s += src[co];
    sup[e] = s;
  }
  __syncthreads();
  for (int it = 0; it < 64; ++it) {
    const int g  = t + it * 256;              // rq<<12 | rk<<10 | di<<8 | dj<<6 | c
    const int c  = g & 63, dj = (g >> 6) & 3, di = (g >> 8) & 3;
    const int rk = (g >> 10) & 3, rq = (g >> 12) & 3;
    const float* wrow = Wd + ((di * 4 + dj) * 64 + c) * 64;
    const float* srow = sup + ((3 - rq) * 4 + (3 - rk)) * 64;
    float s = 0.f;
    for (int ci = 0; ci < 64; ++ci) s += wrow[ci] * srow[ci];
    W2[g] = s * 0.0625f;
  }
  if (t < 16) {
    const int rq = t >> 2, rk = t & 3;
    const float* srow = sup + ((3 - rq) * 4 + (3 - rk)) * 64;
    float s = 0.f;
    for (int ci = 0; ci < 64; ++ci) s += bd[ci] * srow[ci];
    float sb = 0.f;
    for (int co = 0; co < 64; ++co) sb += bu[co];
    cbias[t] = (s + sb) * 0.0625f;
  }
}

// ---------------------------------------------------------------------------
// Kernel B: build the modified queries (global term folded in as identity).
//   Qc[rk][b,p,dj*64+c] = (dj==rk)*q[b,p,c]/16
//                       + sum_di q[b,4*(p/4)+di,c] * W2[p%4,rk,di,dj,c]
// One output element per thread; fully coalesced (c innermost).
// ---------------------------------------------------------------------------
__global__ __launch_bounds__(256) void build_qc(
    const float* __restrict__ q, const float* __restrict__ W2,
    float* __restrict__ qc)
{
  const int id = blockIdx.x * 256 + threadIdx.x;   // rk<<20|b<<16|p<<8|dj<<6|c
  const int c = id & 63, dj = (id >> 6) & 3, p = (id >> 8) & 255;
  const int b = (id >> 16) & 15, rk = (id >> 20) & 3;
  const float* qb = q + b * (L * DK);
  const int i4 = (p >> 2) * 4;
  float val = 0.f;
  #pragma unroll
  for (int di = 0; di < 4; ++di)
    val += qb[(i4 + di) * 64 + c] *
           W2[((((p & 3) * 4 + rk) * 4 + di) * 4 + dj) * 64 + c];
  if (dj == rk) val += qb[p * 64 + c] * 0.0625f;
  qc[id] = val;
}

// ---------------------------------------------------------------------------
// Kernel C: fused logits (WMMA f32) + softmax + attn@V (WMMA f32).
// 256 blocks = 16 batches x 16 query tiles; 128 threads = 4 wave32.
// Wave w owns key columns [64w,64w+64) (= j in [16w,16w+16), all rk) and
// output columns [16w,16w+16).
// ---------------------------------------------------------------------------
__global__ __launch_bounds__(128) void attn_fused(
    const float* __restrict__ k, const float* __restrict__ v,
    const float* __restrict__ qc, const float* __restrict__ cbias,
    float* __restrict__ o_out, float* __restrict__ attn_out)
{
  __shared__ float qcs[16 * 256];   // staged Qc tile for current rk
  __shared__ float sc[16 * 256];    // scores -> probabilities
  __shared__ float red[16 * 8];
  __shared__ float rowm[16];
  __shared__ float rowsum[16];
  __shared__ float cbs[16];

  const int t    = threadIdx.x;
  const int bb   = blockIdx.x >> 4;
  const int p0   = (blockIdx.x & 15) * 16;
  const int lane = t & 31;
  const int w    = t >> 5;
  const int ln   = lane & 15;        // N / M index within fragment
  const int lh   = lane >> 4;        // lane-half selects K pair (ISA 7.12.2)
  const int j0   = w * 16;

  if (t < 16) cbs[t] = cbias[t];
  const float* kb = k + bb * (L * DK);   // Kflat[b]: row j = k rows 4j..4j+3

  for (int rk = 0; rk < 4; ++rk) {
    __syncthreads();
    // stage Qc[rk][bb][p0..p0+15][0..255] (4096 floats) cooperatively
    const v4f* src = (const v4f*)(qc + ((rk * NB + bb) * L + p0) * 256);
    v4f* dst = (v4f*)qcs;
    #pragma unroll
    for (int it = 0; it < 8; ++it) dst[t + it * 128] = src[t + it * 128];
    __syncthreads();

    v8f acc = {0.f, 0.f, 0.f, 0.f, 0.f, 0.f, 0.f, 0.f};
    #pragma unroll 4
    for (int kk = 0; kk < 64; ++kk) {
      const int off = kk * 4 + 2 * lh;                      // K base for lane
      v2f a  = *(const v2f*)(qcs + ln * 256 + off);         // A[M=ln][K=off..]
      v2f bf = *(const v2f*)(kb + (j0 + ln) * 256 + off);   // B[K][N=ln]
      acc = __builtin_amdgcn_wmma_f32_16x16x4_f32(
          false, a, false, bf, (short)0, acc, false, false);
    }
    // scatter tile into score columns r = 4*(j0+n) + rk, plus bias
    #pragma unroll
    for (int vv = 0; vv < 8; ++vv) {
      const int row = vv + 8 * lh;                          // row%4 == vv&3
      sc[row * 256 + 4 * (j0 + ln) + rk] = acc[vv] + cbs[(vv & 3) * 4 + rk];
    }
  }
  __syncthreads();

  // ---- softmax over 256 keys: 8 threads per row, 32 cols each ----
  const int row = t >> 3, sub = t & 7;
  float m = -1e30f;
  for (int c = sub * 32; c < sub * 32 + 32; ++c)
    m = fmaxf(m, sc[row * 256 + c]);
  red[row * 8 + sub] = m;
  __syncthreads();
  if (sub == 0) {
    float mm = red[row * 8];
    #pragma unroll
    for (int s2 = 1; s2 < 8; ++s2) mm = fmaxf(mm, red[row * 8 + s2]);
    rowm[row] = mm;
  }
  __syncthreads();
  const float mm = rowm[row];
  float s = 0.f;
  for (int c = sub * 32; c < sub * 32 + 32; ++c) {
    const float e = __expf(sc[row * 256 + c] - mm);
    sc[row * 256 + c] = e;
    s += e;
  }
  red[row * 8 + sub] = s;
  __syncthreads();
  if (sub == 0) {
    float ss = 0.f;
    #pragma unroll
    for (int s2 = 0; s2 < 8; ++s2) ss += red[row * 8 + s2];
    rowsum[row] = ss;
  }
  __syncthreads();
  const float inv = 1.0f / rowsum[row];
  float* arow = attn_out + (bb * L + p0 + row) * L;
  for (int c = sub * 32; c < sub * 32 + 32; ++c) {
    const float pv = sc[row * 256 + c] * inv;
    sc[row * 256 + c] = pv;
    arow[c] = pv;
  }
  __syncthreads();

  // ---- O(16x64) = P(16x256) @ V(256x64); wave w does columns n0..n0+15 ----
  const float* vb = v + bb * (L * DK);
  const int n0 = j0;
  v8f oacc = {0.f, 0.f, 0.f, 0.f, 0.f, 0.f, 0.f, 0.f};
  #pragma unroll 4
  for (int kk = 0; kk < 64; ++kk) {
    const int kidx = kk * 4 + 2 * lh;
    v2f a = *(const v2f*)(sc + ln * 256 + kidx);            // A = P fragment
    v2f bf;
    bf.x = vb[kidx * 64 + n0 + ln];                         // B[K][N]=V[K][n0+N]
    bf.y = vb[(kidx + 1) * 64 + n0 + ln];
    oacc = __builtin_amdgcn_wmma_f32_16x16x4_f32(
        false, a, false, bf, (short)0, oacc, false, false);
  }
  #pragma unroll
  for (int vv = 0; vv < 8; ++vv)
    o_out[(bb * L + p0 + vv + 8 * lh) * DK + n0 + ln] = oacc[vv];
}

// ---------------------------------------------------------------------------
extern "C" void kernel_launch(void* const* d_in, const int* in_sizes, int n_in,
                              void* d_out, int out_size, void* d_ws, size_t ws_size,
                              hipStream_t stream) {
  const float* q  = (const float*)d_in[0];
  const float* k  = (const float*)d_in[1];
  const float* v  = (const float*)d_in[2];
  const float* Wd = (const float*)d_in[3];
  const float* bd = (const float*)d_in[4];
  const float* Wu = (const float*)d_in[5];
  const float* bu = (const float*)d_in[6];

  float* ws = (float*)d_ws;
  float* W2 = ws;                    // 16384 floats
  float* cb = ws + 16384;            // 16 floats
  float* qc = ws + 32768;            // 4*16*256*256 floats (~16 MB)

  float* o_out    = (float*)d_out;            // [16,256,64]
  float* attn_out = o_out + NB * L * DK;      // [16,256,256]

  prep_weights<<<1, 256, 0, stream>>>(Wd, bd, Wu, bu, W2, cb);
  build_qc<<<16384, 256, 0, stream>>>(q, W2, qc);
  attn_fused<<<256, 128, 0, stream>>>(k, v, qc, cb, o_out, attn_out);
}